// GraphSeparableConv_24421184045264
// MI455X (gfx1250) — compile-verified
//
#include <hip/hip_runtime.h>
#include <hip/hip_bf16.h>

// ---------------------------------------------------------------------------
// GraphSeparableConv (Chebyshev graph conv) for MI455X / gfx1250.
//   N=16, M=49152, Fin=32, Fout=64, K=5, dm=1, E=8*M.
// Pipeline:
//   1) transpose x [N,M,Fin] -> cheb0 [M, N*Fin]
//   2) cheb1 = L@cheb0 (atomic COO spmm); cheb_k = 2 L cheb_{k-1} - cheb_{k-2}
//      (fused: init dst = -cheb_{k-2}, then atomic-add 2*val*src rows)
//   3) pack W[o, k*32+f] = pkernel[o,f]*dkernel[f,0,k] into WMMA A-fragment order
//   4) GEMM W[64,160] @ X[160, N*M] with V_WMMA_F32_16X16X4_F32, + bias + relu
// ---------------------------------------------------------------------------

typedef float v2f __attribute__((ext_vector_type(2)));
typedef float v8f __attribute__((ext_vector_type(8)));

#define MNODES   49152
#define NBATCH   16
#define FIN      32
#define FOUT     64
#define KCH      5
#define NF       (NBATCH * FIN)            // 512
#define PTOT     (NBATCH * MNODES)         // 786432
#define EDGES    (8 * MNODES)              // 393216
#define RDIM     (KCH * FIN)               // 160
#define CHEB_ELEMS ((size_t)MNODES * NF)   // 25,165,824 floats per buffer
#define XS_STRIDE 17                       // 16 points + 1 pad (bank-conflict-free)

// ---------------- 1) transpose x[n,m,f] -> cheb0[m, n*32+f] ----------------
__global__ void k_transpose_x(const float* __restrict__ x, float* __restrict__ c0) {
    size_t t = (size_t)blockIdx.x * blockDim.x + threadIdx.x;   // = n*M*32 + m*32 + f
    int f = (int)(t & 31);
    size_t nm = t >> 5;
    int n = (int)(nm / MNODES);
    int m = (int)(nm % MNODES);
    c0[(size_t)m * NF + (size_t)n * FIN + f] = x[t];
}

// ---------------- zero / negate-copy (float4 granularity) ------------------
__global__ void k_zero4(float4* __restrict__ dst) {
    size_t t = (size_t)blockIdx.x * blockDim.x + threadIdx.x;
    dst[t] = make_float4(0.f, 0.f, 0.f, 0.f);
}
__global__ void k_negcopy4(float4* __restrict__ dst, const float4* __restrict__ src) {
    size_t t = (size_t)blockIdx.x * blockDim.x + threadIdx.x;
    float4 a = src[t];
    dst[t] = make_float4(-a.x, -a.y, -a.z, -a.w);
}

// ---------------- COO SpMM with hardware f32 atomics -----------------------
// One wave (32 lanes) per edge; each lane handles 4 float4 row chunks.
// dst[row, :] += scale * val * src[col, :]
__global__ void k_spmm_atomic(float* __restrict__ dst, const float* __restrict__ src,
                              const float* __restrict__ vals,
                              const int* __restrict__ rows, const int* __restrict__ cols,
                              float scale) {
    int wid  = (int)((blockIdx.x * blockDim.x + threadIdx.x) >> 5);   // edge id
    int lane = (int)(threadIdx.x & 31);
    int row = rows[wid];
    int col = cols[wid];
    float v = vals[wid] * scale;
    const float4* s = (const float4*)(src + (size_t)col * NF);
    float* d = dst + (size_t)row * NF;
#pragma unroll
    for (int it = 0; it < 4; ++it) {
        int c4 = it * 32 + lane;              // coalesced 512B per iter per wave
        float4 a = s[c4];
        int b = c4 * 4;
        unsafeAtomicAdd(d + b + 0, v * a.x);  // global_atomic_add_f32
        unsafeAtomicAdd(d + b + 1, v * a.y);
        unsafeAtomicAdd(d + b + 2, v * a.z);
        unsafeAtomicAdd(d + b + 3, v * a.w);
    }
}

// ---------------- pack combined weights into A-fragment order --------------
// A-fragment (16x4 f32, ISA 7.12.2): lane L(0..15): {K=0,K=1}; lane L+16: {K=2,K=3}.
// apack[((mt*40+kk)*32 + L)*2 + j] = W[mt*16 + (L&15), kk*4 + (L>>4)*2 + j]
__global__ void k_pack_w(const float* __restrict__ pk, const float* __restrict__ dk,
                         float* __restrict__ apack) {
    int t = blockIdx.x * blockDim.x + threadIdx.x;        // 10240 total
    int j = t & 1;
    int L = (t >> 1) & 31;
    int rest = t >> 6;            // mt*40 + kk
    int kk = rest % 40;
    int mt = rest / 40;
    int o = mt * 16 + (L & 15);
    int r = kk * 4 + ((L >> 4) << 1) + j;   // r = k*32 + f
    int k = r >> 5;
    int f = r & 31;
    apack[t] = pk[o * FIN + f] * dk[f * KCH + k];
}

// ---------------- WMMA GEMM: out[p, o] = relu(W @ X + bias) ----------------
// Block = 128 threads = 4 waves; block computes a 64(Fout) x 16(points) tile.
// Wave w handles Fout rows [16w, 16w+16). 40 k-steps of v_wmma_f32_16x16x4_f32.
__global__ void __launch_bounds__(128)
k_gemm_wmma(const float* __restrict__ c0, const float* __restrict__ c1,
            const float* __restrict__ c2, const float* __restrict__ c3,
            const float* __restrict__ c4, const float* __restrict__ apack,
            const float* __restrict__ bias, float* __restrict__ out) {
    __shared__ float Xs[RDIM * XS_STRIDE];   // 160 rows x (16+1) points = 10.6 KB

    const int t    = threadIdx.x;
    const int P0   = blockIdx.x * 16;

    // ---- stage X tile: Xs[k*32+f][p_local] = cheb_k[m*512 + n*32 + f] ----
    {
        const float* cs[KCH] = {c0, c1, c2, c3, c4};
        int f  = t & 31;
        int pi = t >> 5;                    // 0..3
#pragma unroll
        for (int pp = 0; pp < 4; ++pp) {
            int pl = pp * 4 + pi;
            int p  = P0 + pl;
            int n  = p / MNODES;
            int m  = p % MNODES;
            size_t off = (size_t)m * NF + (size_t)n * FIN + f;
#pragma unroll
            for (int k = 0; k < KCH; ++k)
                Xs[((k << 5) + f) * XS_STRIDE + pl] = cs[k][off];
        }
    }
    __syncthreads();

    // ---- WMMA main loop (no divergence: EXEC all ones) ----
    const int lane  = t & 31;
    const int mtile = t >> 5;               // 0..3
    const int hi    = lane >> 4;            // 0: K pair {0,1}, 1: {2,3}
    const int pl    = lane & 15;            // point within tile / N column

    v8f acc = {};
    const v2f* ap = (const v2f*)apack;
#pragma unroll 8
    for (int kk = 0; kk < 40; ++kk) {
        v2f a = ap[(mtile * 40 + kk) * 32 + lane];     // coalesced 256B/wave
        int rk = kk * 4 + hi * 2;
        v2f b;
        b.x = Xs[rk * XS_STRIDE + pl];
        b.y = Xs[(rk + 1) * XS_STRIDE + pl];
        acc = __builtin_amdgcn_wmma_f32_16x16x4_f32(
            /*neg_a=*/false, a, /*neg_b=*/false, b,
            /*c_mod=*/(short)0, acc, /*reuse_a=*/false, /*reuse_b=*/false);
    }

    // ---- epilogue: bias + relu; C/D layout: VGPR v -> row v (lanes 0-15) / v+8 ----
    const int p      = P0 + pl;
    const int base_o = mtile * 16 + hi * 8;
    float4 o0, o1;
    o0.x = fmaxf(acc[0] + bias[base_o + 0], 0.f);
    o0.y = fmaxf(acc[1] + bias[base_o + 1], 0.f);
    o0.z = fmaxf(acc[2] + bias[base_o + 2], 0.f);
    o0.w = fmaxf(acc[3] + bias[base_o + 3], 0.f);
    o1.x = fmaxf(acc[4] + bias[base_o + 4], 0.f);
    o1.y = fmaxf(acc[5] + bias[base_o + 5], 0.f);
    o1.z = fmaxf(acc[6] + bias[base_o + 6], 0.f);
    o1.w = fmaxf(acc[7] + bias[base_o + 7], 0.f);
    float4* op = (float4*)(out + (size_t)p * FOUT + base_o);
    op[0] = o0;
    op[1] = o1;
}

// ---------------------------------------------------------------------------
extern "C" void kernel_launch(void* const* d_in, const int* in_sizes, int n_in,
                              void* d_out, int out_size, void* d_ws, size_t ws_size,
                              hipStream_t stream) {
    const float* x     = (const float*)d_in[0];
    const float* evals = (const float*)d_in[1];
    const float* dk    = (const float*)d_in[2];
    const float* pk    = (const float*)d_in[3];
    const float* bias  = (const float*)d_in[4];
    const int*   erow  = (const int*)d_in[5];
    const int*   ecol  = (const int*)d_in[6];
    float* out = (float*)d_out;

    float* ws = (float*)d_ws;
    float* c0 = ws;
    float* c1 = c0 + CHEB_ELEMS;
    float* c2 = c1 + CHEB_ELEMS;
    float* c3 = c2 + CHEB_ELEMS;
    float* c4 = c3 + CHEB_ELEMS;
    float* apack = c4 + CHEB_ELEMS;          // 10240 floats

    const int elems4      = (int)(CHEB_ELEMS / 4);       // 6,291,456
    const int gridCheb    = (int)(CHEB_ELEMS / 256);     // 98,304
    const int gridCheb4   = elems4 / 256;                // 24,576
    const int gridSpmm    = (EDGES * 32) / 256;          // 49,152
    const int gridGemm    = PTOT / 16;                   // 49,152

    // 1) x -> cheb0 (row-major [M, N*Fin])
    k_transpose_x<<<gridCheb, 256, 0, stream>>>(x, c0);

    // 2) Chebyshev recursion with fused init + atomic SpMM
    k_zero4<<<gridCheb4, 256, 0, stream>>>((float4*)c1);
    k_spmm_atomic<<<gridSpmm, 256, 0, stream>>>(c1, c0, evals, erow, ecol, 1.0f);

    k_negcopy4<<<gridCheb4, 256, 0, stream>>>((float4*)c2, (const float4*)c0);
    k_spmm_atomic<<<gridSpmm, 256, 0, stream>>>(c2, c1, evals, erow, ecol, 2.0f);

    k_negcopy4<<<gridCheb4, 256, 0, stream>>>((float4*)c3, (const float4*)c1);
    k_spmm_atomic<<<gridSpmm, 256, 0, stream>>>(c3, c2, evals, erow, ecol, 2.0f);

    k_negcopy4<<<gridCheb4, 256, 0, stream>>>((float4*)c4, (const float4*)c2);
    k_spmm_atomic<<<gridSpmm, 256, 0, stream>>>(c4, c3, evals, erow, ecol, 2.0f);

    // 3) pack combined weights into WMMA A-fragment layout
    k_pack_w<<<40, 256, 0, stream>>>(pk, dk, apack);

    // 4) fused GEMM + bias + relu via V_WMMA_F32_16X16X4_F32
    k_gemm_wmma<<<gridGemm, 128, 0, stream>>>(c0, c1, c2, c3, c4, apack, bias, out);
}